// RobustSigmaDistance_10625749090598
// MI455X (gfx1250) — compile-verified
//
#include <hip/hip_runtime.h>
#include <hip/hip_bf16.h>
#include <math.h>

typedef __attribute__((ext_vector_type(2))) float v2f;
typedef __attribute__((ext_vector_type(8))) float v8f;

#define NPTS   4096
#define BATCH  16
#define TILE   16
#define NTILES (NPTS / TILE)          // 256 candidate tiles
#define QWAVES 8                      // 8 waves -> 128 queries per block
#define QPB    (QWAVES * TILE)        // queries per block
#define NELEM  (NPTS * 3)             // 12288 residual scalars per (b,dir)
#define SORTN  16384                  // pow2 pad for bitonic

// ---------------------------------------------------------------------------
// Kernel 1: brute-force NN via V_WMMA_F32_16X16X4_F32, emit signed residuals.
//
// Score: D[m,n] = |c_m|^2 - 2 c_m.q_n + (|q_n|^2 + 1) = d^2(m,n) + 1  >= 1.
// The +|q|^2+1 term rides in through the WMMA C accumulator (per-lane splat,
// lane == query column). Positive f32 bit patterns are order-isomorphic to
// u32, so argmin becomes a u32 min over packed keys.
//
// Key scheme (2 VALU/element): tile-local key = (bits & 0xFFFFF000) | r
// (r = 0..7, OR with small immediate), v_min3_u32-tree the 8 locals, then a
// single OR attaches the tile base (multiple of 8, disjoint from r) before
// folding into the running best. Ties resolve to lower index like jnp.argmin.
//
// A (candidates, 16x4): lane L holds row m=L%16; VGPR0 = K0/K2, VGPR1 = K1/K3
//   -> 8-byte LDS read of (-2x,-2y) [lanes<16] or (-2z,|c|^2) [lanes>=16].
// B (queries, 4x16): lane L holds col n=L%16; (qx,qy) / (qz,1). Loop-constant.
// D (16x16 f32): lane = column n; VGPR r = row r (lanes<16) / r+8 (lanes>=16).
// A-operand LDS loads are software-pipelined one tile ahead of the WMMA.
// ---------------------------------------------------------------------------
__global__ __launch_bounds__(256) void nn_resid_kernel(
    const float* __restrict__ X, const float* __restrict__ Y,
    float* __restrict__ resid)
{
    __shared__ float ldsC[NPTS * 4];   // (-2x, -2y, -2z, |c|^2): 64 KB

    const int qg  = blockIdx.x & (NPTS / QPB - 1);           // query group (0..31)
    const int dir = (blockIdx.x >> 5) & 1;                   // 0: X->Y, 1: Y->X
    const int b   = blockIdx.x >> 6;                         // batch

    const float* __restrict__ Q = (dir == 0 ? X : Y) + (size_t)b * NPTS * 3;
    const float* __restrict__ C = (dir == 0 ? Y : X) + (size_t)b * NPTS * 3;

    // Stage transformed candidates into LDS (whole block cooperates).
    for (int i = threadIdx.x; i < NPTS; i += blockDim.x) {
        float cx = C[i * 3 + 0], cy = C[i * 3 + 1], cz = C[i * 3 + 2];
        float4 v = make_float4(-2.0f * cx, -2.0f * cy, -2.0f * cz,
                               cx * cx + cy * cy + cz * cz);
        reinterpret_cast<float4*>(ldsC)[i] = v;
    }
    __syncthreads();

    const int wave = threadIdx.x >> 5;
    const int lane = threadIdx.x & 31;
    const int n    = lane & 15;          // query column within tile
    const bool hi  = lane >= 16;         // K=2,3 half / rows 8..15 half

    const int   qidx = qg * QPB + wave * TILE + n;
    const float qx = Q[qidx * 3 + 0];
    const float qy = Q[qidx * 3 + 1];
    const float qz = Q[qidx * 3 + 2];
    const float qbias = qx * qx + qy * qy + qz * qz + 1.0f;   // |q|^2 + 1

    v2f bop;                              // B operand: (x,y) or (z,1)
    bop.x = hi ? qz   : qx;
    bop.y = hi ? 1.0f : qy;

    const int laneAOff = n * 4 + (hi ? 2 : 0);   // LDS float offset within a tile row
    const unsigned int rowHalf = hi ? 8u : 0u;

    unsigned int bestKey = 0xFFFFFFFFu;

    // Software pipeline: prefetch A operand for tile 0.
    v2f aop = *reinterpret_cast<const v2f*>(&ldsC[laneAOff]);

    for (int t = 0; t < NTILES; ++t) {
        // Prefetch next tile's A operand (overlaps with WMMA + key reduce).
        v2f anext = aop;
        if (t + 1 < NTILES)
            anext = *reinterpret_cast<const v2f*>(&ldsC[(t + 1) * TILE * 4 + laneAOff]);

        v8f acc;
#pragma unroll
        for (int r = 0; r < 8; ++r) acc[r] = qbias;   // C = |q|^2 + 1 splat

        acc = __builtin_amdgcn_wmma_f32_16x16x4_f32(
            /*neg_a=*/false, aop, /*neg_b=*/false, bop,
            /*c_mod=*/(short)0, acc, /*reuse_a=*/false, /*reuse_b=*/false);

        // Tile-local packed keys: value bits | row-within-half (0..7).
        unsigned int k[8];
#pragma unroll
        for (int r = 0; r < 8; ++r)
            k[r] = (__float_as_uint(acc[r]) & 0xFFFFF000u) | (unsigned)r;

        // 8 -> 1 via min3 tree (local keys only, no per-value index adds).
        unsigned int m = min(min(k[0], k[1]), min(k[2], k[3]));
        m = min(m, min(min(k[4], k[5]), min(k[6], k[7])));

        // Attach tile base once (disjoint bits: base multiple of 8, r < 8).
        const unsigned int base = (unsigned int)(t * TILE) | rowHalf;
        bestKey = min(bestKey, m | base);

        aop = anext;
    }

    // Merge the two lane-halves (rows 0-7 vs 8-15 of every tile).
    bestKey = min(bestKey, (unsigned int)__shfl_xor((int)bestKey, 16, 32));

    if (lane < 16) {
        const int bestIdx = (int)(bestKey & 0xFFFu);
        const float* c = &C[bestIdx * 3];
        float* out = &resid[(((size_t)(b * 2 + dir)) * NPTS + qidx) * 3];
        out[0] = qx - c[0];
        out[1] = qy - c[1];
        out[2] = qz - c[2];
    }
}

// ---------------------------------------------------------------------------
// Kernel 2: robust masked std per (batch, direction).
// Bitonic sort 12288 residuals (padded to 16384 with +inf) in LDS, take
// interpolated 0.15/0.85 quantiles, build mask with fallback chain, then
// weighted unbiased std. One 512-thread block per (b,dir).
// ---------------------------------------------------------------------------
__global__ __launch_bounds__(512) void robust_sigma_kernel(
    const float* __restrict__ resid, float* __restrict__ sigma)
{
    __shared__ float s[SORTN];   // 64 KB; tail [NELEM..SORTN) reused as scratch

    const int tid = threadIdx.x;
    const int nt  = blockDim.x;
    const int lane = tid & 31;
    const float* __restrict__ x = resid + (size_t)blockIdx.x * NELEM;

    for (int i = tid; i < SORTN; i += nt)
        s[i] = (i < NELEM) ? x[i] : INFINITY;
    __syncthreads();

    // Bitonic sort ascending.
    for (int k = 2; k <= SORTN; k <<= 1) {
        for (int j = k >> 1; j > 0; j >>= 1) {
            for (int i = tid; i < SORTN; i += nt) {
                int ixj = i ^ j;
                if (ixj > i) {
                    float a = s[i], b = s[ixj];
                    bool up = ((i & k) == 0);
                    if (up ? (a > b) : (a < b)) { s[i] = b; s[ixj] = a; }
                }
            }
            __syncthreads();
        }
    }

    // Interpolated quantiles (everyone computes; values are in LDS).
    const float p_lo = 0.15f * (float)(NELEM - 1);
    const float p_hi = 0.85f * (float)(NELEM - 1);
    const int   ilo = (int)p_lo, ihi = (int)p_hi;
    const float qlo = s[ilo] + (p_lo - (float)ilo) * (s[ilo + 1] - s[ilo]);
    const float qhi = s[ihi] + (p_hi - (float)ihi) * (s[ihi + 1] - s[ihi]);

    float* scr = &s[NELEM];      // padding region, free after sort
    if (tid < 8) scr[tid] = 0.0f;
    __syncthreads();

    // Pass 1: sum, sumsq, quantile-mask count.
    float sum = 0.0f, sq = 0.0f, mcnt = 0.0f;
    for (int i = tid; i < NELEM; i += nt) {
        float v = s[i];
        sum += v; sq += v * v;
        if (v < qlo || v > qhi) mcnt += 1.0f;
    }
#pragma unroll
    for (int o = 16; o > 0; o >>= 1) {
        sum  += __shfl_down(sum, o, 32);
        sq   += __shfl_down(sq, o, 32);
        mcnt += __shfl_down(mcnt, o, 32);
    }
    if (lane == 0) {
        atomicAdd(&scr[0], sum);
        atomicAdd(&scr[1], sq);
        atomicAdd(&scr[2], mcnt);
    }
    __syncthreads();

    const float nf   = (float)NELEM;
    const float mu   = scr[0] / nf;
    const float sd   = sqrtf(fmaxf(0.0f, (scr[1] - nf * mu * mu) / (nf - 1.0f)));
    const float mCnt = scr[2];

    // Pass 2: simple-mask count (|x-mu| > 1*sd).
    float scnt = 0.0f;
    for (int i = tid; i < NELEM; i += nt)
        if (fabsf(s[i] - mu) > sd) scnt += 1.0f;
#pragma unroll
    for (int o = 16; o > 0; o >>= 1) scnt += __shfl_down(scnt, o, 32);
    if (lane == 0) atomicAdd(&scr[3], scnt);
    __syncthreads();

    const float sCnt = scr[3];
    const int mode = (mCnt > 0.0f) ? 0 : ((sCnt > 0.0f) ? 1 : 2);
    const float cnt = (mode == 0) ? mCnt : ((mode == 1) ? sCnt : nf);

    // Pass 3: weighted sum with the chosen mask.
    float wsum = 0.0f;
    for (int i = tid; i < NELEM; i += nt) {
        float v = s[i];
        bool w = (mode == 0) ? (v < qlo || v > qhi)
               : (mode == 1) ? (fabsf(v - mu) > sd) : true;
        if (w) wsum += v;
    }
#pragma unroll
    for (int o = 16; o > 0; o >>= 1) wsum += __shfl_down(wsum, o, 32);
    if (lane == 0) atomicAdd(&scr[4], wsum);
    __syncthreads();

    const float meanw = scr[4] / cnt;

    // Pass 4: weighted variance (unbiased).
    float wvar = 0.0f;
    for (int i = tid; i < NELEM; i += nt) {
        float v = s[i];
        bool w = (mode == 0) ? (v < qlo || v > qhi)
               : (mode == 1) ? (fabsf(v - mu) > sd) : true;
        if (w) { float d = v - meanw; wvar += d * d; }
    }
#pragma unroll
    for (int o = 16; o > 0; o >>= 1) wvar += __shfl_down(wvar, o, 32);
    if (lane == 0) atomicAdd(&scr[5], wvar);
    __syncthreads();

    if (tid == 0)
        sigma[blockIdx.x] = sqrtf(fmaxf(0.0f, scr[5] / (cnt - 1.0f)));
}

// ---------------------------------------------------------------------------
// Kernel 3: out = mean_b max(sigma[2b], sigma[2b+1]).
// ---------------------------------------------------------------------------
__global__ void finalize_kernel(const float* __restrict__ sigma,
                                float* __restrict__ out)
{
    if (threadIdx.x == 0) {
        float acc = 0.0f;
        for (int b = 0; b < BATCH; ++b)
            acc += fmaxf(sigma[2 * b], sigma[2 * b + 1]);
        *out = acc / (float)BATCH;
    }
}

extern "C" void kernel_launch(void* const* d_in, const int* in_sizes, int n_in,
                              void* d_out, int out_size, void* d_ws, size_t ws_size,
                              hipStream_t stream)
{
    (void)in_sizes; (void)n_in; (void)out_size; (void)ws_size;
    const float* x = (const float*)d_in[0];   // [16,4096,3]
    const float* y = (const float*)d_in[1];   // [16,4096,3]

    float* resid = (float*)d_ws;                         // [32][4096][3] = 1.5 MB
    float* sigma = resid + (size_t)BATCH * 2 * NPTS * 3; // [32]

    // B * 2 directions * (4096/128) query groups = 1024 blocks
    nn_resid_kernel<<<BATCH * 2 * (NPTS / QPB), 256, 0, stream>>>(x, y, resid);
    robust_sigma_kernel<<<BATCH * 2, 512, 0, stream>>>(resid, sigma);
    finalize_kernel<<<1, 32, 0, stream>>>(sigma, (float*)d_out);
}